// TransR_74938589381291
// MI455X (gfx1250) — compile-verified
//
#include <hip/hip_runtime.h>
#include <hip/hip_bf16.h>

#define D_DIM   128
#define N_REL   1000
#define BATCH   16384
#define MAX_TILES 2048   // worst case: sum ceil(c_r/16) <= 1024 + 1000 = 2024

typedef __attribute__((ext_vector_type(2))) float v2f;
typedef __attribute__((ext_vector_type(8))) float v8f;

// ---------------------------------------------------------------- k_init
__global__ void k_init(int* __restrict__ counts, int* __restrict__ tileCount) {
    int i = blockIdx.x * blockDim.x + threadIdx.x;
    if (i < N_REL) counts[i] = 0;
    if (i == 0) *tileCount = 0;
}

// ------------------------------------------------- k_xdiff: x = h/|h| - t/|t|
__global__ __launch_bounds__(D_DIM)
void k_xdiff(const float* __restrict__ h, const float* __restrict__ t,
             const int* __restrict__ idx, float* __restrict__ x,
             int* __restrict__ counts) {
    int b = blockIdx.x;
    int r = threadIdx.x;
    float hv = h[b * D_DIM + r];
    float tv = t[b * D_DIM + r];
    __shared__ float sh[D_DIM];
    __shared__ float st[D_DIM];
    sh[r] = hv * hv;
    st[r] = tv * tv;
    __syncthreads();
    for (int s = D_DIM / 2; s > 0; s >>= 1) {
        if (r < s) { sh[r] += sh[r + s]; st[r] += st[r + s]; }
        __syncthreads();
    }
    float nh = fmaxf(sqrtf(sh[0]), 1e-12f);
    float nt = fmaxf(sqrtf(st[0]), 1e-12f);
    x[b * D_DIM + r] = hv / nh - tv / nt;
    if (r == 0) atomicAdd(&counts[idx[b]], 1);
}

// ------------------------ k_scan: offsets, cursors, tile list (one block)
__global__ __launch_bounds__(1024)
void k_scan(const int* __restrict__ counts, int* __restrict__ offsets,
            int* __restrict__ cursors, int4* __restrict__ tiles,
            int* __restrict__ tileCount) {
    __shared__ int buf[1024];
    int t = threadIdx.x;
    int c = (t < N_REL) ? counts[t] : 0;
    buf[t] = c;
    __syncthreads();
    for (int off = 1; off < 1024; off <<= 1) {
        int v = (t >= off) ? buf[t - off] : 0;
        __syncthreads();
        buf[t] += v;
        __syncthreads();
    }
    int incl = buf[t];
    int excl = incl - c;
    if (t < N_REL) { offsets[t] = excl; cursors[t] = excl; }
    if (t == N_REL - 1) offsets[N_REL] = incl;

    // second scan: tiles per relation
    int nt = (t < N_REL) ? ((c + 15) >> 4) : 0;
    __syncthreads();
    buf[t] = nt;
    __syncthreads();
    for (int off = 1; off < 1024; off <<= 1) {
        int v = (t >= off) ? buf[t - off] : 0;
        __syncthreads();
        buf[t] += v;
        __syncthreads();
    }
    int tExcl = buf[t] - nt;
    for (int i = 0; i < nt; ++i) {
        int cnt = min(16, c - 16 * i);
        tiles[tExcl + i] = make_int4(t, excl + 16 * i, cnt, 0);
    }
    if (t == 1023) *tileCount = buf[1023];
}

// ---------------------------------------------------------------- k_scatter
__global__ void k_scatter(const int* __restrict__ idx, int* __restrict__ cursors,
                          int* __restrict__ order) {
    int b = blockIdx.x * blockDim.x + threadIdx.x;
    if (b < BATCH) {
        int p = atomicAdd(&cursors[idx[b]], 1);
        order[p] = b;
    }
}

// --------------- k_gemm: per tile, Y = P_rel * X + E via WMMA f32 16x16x4
// then score[col] = -sum_rows (Y^2). 128 threads = 4 waves; 8 row-tiles of Y,
// 2 per wave. A fragment (16x4 f32): lanes 0-15 K=0,1 ; lanes 16-31 K=2,3.
// C/D (16x16 f32): VGPR j -> row j + 8*(lane>=16), col = lane&15.
__global__ __launch_bounds__(128)
void k_gemm(const float* __restrict__ proj, const float* __restrict__ edge,
            const float* __restrict__ xw, const int* __restrict__ order,
            const int4* __restrict__ tiles, const int* __restrict__ tileCount,
            float* __restrict__ out) {
    int tile = blockIdx.x;
    if (tile >= *tileCount) return;
    int4 td = tiles[tile];
    int rel = td.x, start = td.y, cnt = td.z;

    __shared__ float xs[D_DIM][17];   // X tile, row-major: xs[k][col]
    __shared__ float es[D_DIM][17];   // E tile, row-major: es[row][col]
    __shared__ float part[8][16];     // per-row-tile per-column sum of squares

    int tid = threadIdx.x;

    // stage X and E (coalesced per column: 128 consecutive floats)
    for (int c = 0; c < 16; ++c) {
        int b = (c < cnt) ? order[start + c] : -1;
        float xv = 0.0f, ev = 0.0f;
        if (b >= 0) {
            xv = xw[b * D_DIM + tid];
            ev = edge[b * D_DIM + tid];
        }
        xs[tid][c] = xv;
        es[tid][c] = ev;
    }
    __syncthreads();

    int wave = tid >> 5;
    int lane = tid & 31;
    int half = lane >> 4;   // 0: K=0,1  1: K=2,3
    int m    = lane & 15;

    const float* Pbase = proj + (size_t)rel * (D_DIM * D_DIM);

    for (int rtp = 0; rtp < 2; ++rtp) {
        int rt = wave + 4 * rtp;                 // row-tile 0..7
        const float* Prow = Pbase + (rt * 16 + m) * D_DIM + 2 * half;

        // init accumulator with E tile -> computes P*x + e directly
        v8f acc;
#pragma unroll
        for (int j = 0; j < 8; ++j) acc[j] = es[rt * 16 + 8 * half + j][m];

#pragma unroll 4
        for (int k = 0; k < D_DIM; k += 4) {
            v2f A;  A[0] = Prow[k];               A[1] = Prow[k + 1];
            v2f Bv; Bv[0] = xs[k + 2 * half][m];  Bv[1] = xs[k + 2 * half + 1][m];
            acc = __builtin_amdgcn_wmma_f32_16x16x4_f32(
                false, A, false, Bv, (short)0, acc, false, false);
        }

        // sum of squares over this lane's 8 rows, combine halves
        float p = 0.0f;
#pragma unroll
        for (int j = 0; j < 8; ++j) p += acc[j] * acc[j];
        p += __shfl_xor(p, 16, 32);
        if (half == 0) part[rt][m] = p;
    }
    __syncthreads();

    // fixed-order combine across the 8 row-tiles -> deterministic
    if (tid < 16 && tid < cnt) {
        float s = 0.0f;
#pragma unroll
        for (int r = 0; r < 8; ++r) s += part[r][tid];
        out[order[start + tid]] = -s;
    }
}

// ---------------------------------------------------------------- launcher
extern "C" void kernel_launch(void* const* d_in, const int* in_sizes, int n_in,
                              void* d_out, int out_size, void* d_ws, size_t ws_size,
                              hipStream_t stream) {
    const float* head = (const float*)d_in[0];
    const float* tail = (const float*)d_in[1];
    const float* edge = (const float*)d_in[2];
    const float* proj = (const float*)d_in[3];
    const int*   eidx = (const int*)d_in[4];
    float* out = (float*)d_out;

    // workspace layout
    char* ws = (char*)d_ws;
    float* xw      = (float*)ws;                               // BATCH*128 f32 = 8 MB
    int*   counts  = (int*)(ws + (size_t)BATCH * D_DIM * 4);   // 1024 ints
    int*   offsets = counts + 1024;                            // 1024 ints (1001 used)
    int*   cursors = offsets + 1024;                           // 1024 ints
    int*   order   = cursors + 1024;                           // BATCH ints
    int4*  tiles   = (int4*)(order + BATCH);                   // MAX_TILES int4
    int*   tileCnt = (int*)(tiles + MAX_TILES);

    k_init   <<<4, 256, 0, stream>>>(counts, tileCnt);
    k_xdiff  <<<BATCH, D_DIM, 0, stream>>>(head, tail, eidx, xw, counts);
    k_scan   <<<1, 1024, 0, stream>>>(counts, offsets, cursors, tiles, tileCnt);
    k_scatter<<<(BATCH + 255) / 256, 256, 0, stream>>>(eidx, cursors, order);
    k_gemm   <<<MAX_TILES, 128, 0, stream>>>(proj, edge, xw, order, tiles, tileCnt, out);
}